// UnetGNNLayer_84198538871361
// MI455X (gfx1250) — compile-verified
//
#include <hip/hip_runtime.h>

typedef float v2f __attribute__((ext_vector_type(2)));
typedef float v8f __attribute__((ext_vector_type(8)));
typedef int   i4  __attribute__((vector_size(16)));
typedef __attribute__((address_space(1))) i4 gi4;   // global int4
typedef __attribute__((address_space(3))) i4 li4;   // LDS int4

#define NCH 256            // channels C
#define HEIGHT 60
#define KDIM 512           // contraction dim after fusing (x_dst ; x_src)
#define XSTRIDE 72         // LDS row stride (floats): 64 data + pad for bank spread
#define TILE_CNT (NCH * HEIGHT)   // 15360 elements per edge tile

// ---------------- CDNA5 async staging helpers ----------------

__device__ __forceinline__ void async_wait0() {
#if __has_builtin(__builtin_amdgcn_s_wait_asynccnt)
  __builtin_amdgcn_s_wait_asynccnt(0);
#else
  asm volatile("s_wait_asynccnt 0x0" ::: "memory");
#endif
}

// ---------------- prep kernels ----------------

__global__ __launch_bounds__(256) void zero_kernel(float* __restrict__ p, int n) {
  int i = blockIdx.x * 256 + threadIdx.x;
  if (i < n) p[i] = 0.f;
}

// Wf[o][k] : k<256 -> W1-W2 ; k>=256 -> W2   (W_conv is (256, 768) row-major)
__global__ __launch_bounds__(256) void prep_wf(const float* __restrict__ Wc, float* __restrict__ Wf) {
  int i = blockIdx.x * 256 + threadIdx.x;
  if (i >= NCH * KDIM) return;
  int o = i >> 9, k = i & 511;
  float v = Wc[o * 768 + k];
  if (k < 256) v -= Wc[o * 768 + k + 256];
  Wf[i] = v;
}

// ea_lin[e][c] = b_edge[c] + sum_k edge_attr[e,k] * W_edge[k,c]
__global__ __launch_bounds__(256) void prep_ea(const float* __restrict__ ea,
                                               const float* __restrict__ We,
                                               const float* __restrict__ be,
                                               float* __restrict__ out, int E) {
  int i = blockIdx.x * 256 + threadIdx.x;
  if (i >= E * NCH) return;
  int e = i >> 8, c = i & 255;
  float s = be[c];
#pragma unroll
  for (int k = 0; k < 5; ++k) s += ea[e * 5 + k] * We[k * NCH + c];
  out[i] = s;
}

// v3[e][o] = sum_c ea_lin[e,c] * W_conv[o, 512+c]
__global__ __launch_bounds__(256) void prep_v3(const float* __restrict__ ea_lin,
                                               const float* __restrict__ Wc,
                                               float* __restrict__ v3, int E) {
  int i = blockIdx.x * 256 + threadIdx.x;
  if (i >= E * NCH) return;
  int e = i >> 8, o = i & 255;
  const float* a = ea_lin + (size_t)e * NCH;
  const float* w = Wc + (size_t)o * 768 + 512;
  float s = 0.f;
#pragma unroll 8
  for (int c = 0; c < NCH; ++c) s += a[c] * w[c];
  v3[i] = s;
}

// ---------------- main fused edge kernel ----------------
// One workgroup (8 wave32) per edge: (256x512)@(512x60) fp32 WMMA GEMM
// + LayerNorm(256x60) + ReLU + atomic scatter-add into out[dst].

__global__ __launch_bounds__(256) void edge_gemm_ln_scatter(
    const float* __restrict__ x,       // (384, 256, 60)
    const int*   __restrict__ ei,      // (2, E)
    const float* __restrict__ v3,      // (E, 256)
    const float* __restrict__ Wf,      // (256, 512) fused weights
    const float* __restrict__ gam,     // (256, 60)
    const float* __restrict__ bet,     // (256, 60)
    float* __restrict__ out,           // (384, 256, 60)
    int E) {
  __shared__ float Xl[KDIM * XSTRIDE];   // 144 KB: [x_dst ; x_src] as 512 x 64 (padded)
  __shared__ float v3s[NCH];
  __shared__ float red[2];

  const int e = blockIdx.x;
  const int t = threadIdx.x;
  const int src = ei[e];
  const int dst = ei[E + e];

  if (t == 0) { red[0] = 0.f; red[1] = 0.f; }
  v3s[t] = v3[(size_t)e * NCH + t];

  // Warm L2/L0 for this wave's weight stripe (Wf is shared by all blocks).
  __builtin_prefetch(Wf + (size_t)t * KDIM, 0, 3);

  // Stage B matrix: row c (<256) = x[dst][c][:], row 256+c = x[src][c][:]
  // CDNA5 async copy straight into LDS (tracked by ASYNCcnt).
  {
    const float* xd = x + (size_t)dst * TILE_CNT + t * HEIGHT;
    const float* xs = x + (size_t)src * TILE_CNT + t * HEIGHT;
    float* ld = Xl + t * XSTRIDE;
    float* ls = Xl + (NCH + t) * XSTRIDE;
#if __has_builtin(__builtin_amdgcn_global_load_async_to_lds_b128)
#pragma unroll
    for (int j = 0; j < 15; ++j) {
      __builtin_amdgcn_global_load_async_to_lds_b128(
          (gi4*)(xd + 4 * j), (li4*)(ld + 4 * j), 0, 0);
      __builtin_amdgcn_global_load_async_to_lds_b128(
          (gi4*)(xs + 4 * j), (li4*)(ls + 4 * j), 0, 0);
    }
#else
#pragma unroll
    for (int j = 0; j < 15; ++j) {
      ((float4*)ld)[j] = ((const float4*)xd)[j];
      ((float4*)ls)[j] = ((const float4*)xs)[j];
    }
#endif
    float4 z; z.x = z.y = z.z = z.w = 0.f;
    ((float4*)ld)[15] = z;           // zero pad columns 60..63
    ((float4*)ls)[15] = z;
  }
  async_wait0();
  __syncthreads();

  const int lane = t & 31;
  const int wv   = t >> 5;
  const int hi   = lane >> 4;        // half-wave select (K pairs / M+8 rows)
  const int l16  = lane & 15;
  const int m0   = wv * 32;          // this wave's M stripe

  // Accumulators: 2 M-tiles x 4 N-tiles of 16x16 f32, C preloaded with v3 bias
  v8f acc[2][4];
#pragma unroll
  for (int mt = 0; mt < 2; ++mt) {
#pragma unroll
    for (int r = 0; r < 8; ++r) {
      float cv = v3s[m0 + mt * 16 + hi * 8 + r];
      acc[mt][0][r] = cv; acc[mt][1][r] = cv;
      acc[mt][2][r] = cv; acc[mt][3][r] = cv;
    }
  }

  const float* w0 = Wf + (size_t)(m0 + l16) * KDIM;   // A rows for M-tile 0
  const float* w1 = w0 + 16 * KDIM;                   // A rows for M-tile 1

#pragma unroll 4
  for (int k0 = 0; k0 < KDIM; k0 += 4) {
    const int kb = k0 + 2 * hi;                       // ISA 16x4 A/B fragment layout
    v2f a0 = *(const v2f*)(w0 + kb);
    v2f a1 = *(const v2f*)(w1 + kb);
    const float* xr0 = Xl + kb * XSTRIDE + l16;
    const float* xr1 = xr0 + XSTRIDE;
#pragma unroll
    for (int nt = 0; nt < 4; ++nt) {
      v2f bb;
      bb.x = xr0[nt * 16];
      bb.y = xr1[nt * 16];
      acc[0][nt] = __builtin_amdgcn_wmma_f32_16x16x4_f32(
          false, a0, false, bb, (short)0, acc[0][nt], false, false);
      acc[1][nt] = __builtin_amdgcn_wmma_f32_16x16x4_f32(
          false, a1, false, bb, (short)0, acc[1][nt], false, false);
    }
  }

  // ---- LayerNorm statistics over the valid 256x60 tile ----
  float s = 0.f, s2 = 0.f;
#pragma unroll
  for (int nt = 0; nt < 4; ++nt) {
    int n = nt * 16 + l16;
    if (n < HEIGHT) {
#pragma unroll
      for (int mt = 0; mt < 2; ++mt)
#pragma unroll
        for (int r = 0; r < 8; ++r) { float v = acc[mt][nt][r]; s += v; s2 += v * v; }
    }
  }
#pragma unroll
  for (int off = 16; off > 0; off >>= 1) {
    s  += __shfl_xor(s,  off, 32);
    s2 += __shfl_xor(s2, off, 32);
  }
  if (lane == 0) { atomicAdd(&red[0], s); atomicAdd(&red[1], s2); }
  __syncthreads();

  const float inv = 1.f / (float)TILE_CNT;
  const float mu  = red[0] * inv;
  const float var = red[1] * inv - mu * mu;
  const float rs  = rsqrtf(var + 1e-5f);

  // ---- normalize + affine + ReLU + scatter-add ----
  float* op = out + (size_t)dst * TILE_CNT;
#pragma unroll
  for (int nt = 0; nt < 4; ++nt) {
    int n = nt * 16 + l16;
    if (n < HEIGHT) {
#pragma unroll
      for (int mt = 0; mt < 2; ++mt) {
#pragma unroll
        for (int r = 0; r < 8; ++r) {
          int m = m0 + mt * 16 + hi * 8 + r;
          int idx = m * HEIGHT + n;
          float v = (acc[mt][nt][r] - mu) * rs * gam[idx] + bet[idx];
          v = v > 0.f ? v : 0.f;
          unsafeAtomicAdd(op + idx, v);
        }
      }
    }
  }
}

// ---------------- launcher ----------------

extern "C" void kernel_launch(void* const* d_in, const int* in_sizes, int n_in,
                              void* d_out, int out_size, void* d_ws, size_t ws_size,
                              hipStream_t stream) {
  const float* x         = (const float*)d_in[0];
  const float* edge_attr = (const float*)d_in[1];
  const int*   edge_idx  = (const int*)d_in[2];
  const float* W_edge    = (const float*)d_in[3];
  const float* b_edge    = (const float*)d_in[4];
  const float* W_conv    = (const float*)d_in[5];
  const float* ln_gamma  = (const float*)d_in[6];
  const float* ln_beta   = (const float*)d_in[7];
  float* out = (float*)d_out;

  const int E = in_sizes[2] / 2;   // 3072

  float* ws     = (float*)d_ws;
  float* ea_lin = ws;                          // E*256
  float* v3     = ea_lin + (size_t)E * NCH;    // E*256
  float* Wf     = v3 + (size_t)E * NCH;        // 256*512

  zero_kernel<<<(out_size + 255) / 256, 256, 0, stream>>>(out, out_size);
  prep_wf<<<(NCH * KDIM + 255) / 256, 256, 0, stream>>>(W_conv, Wf);
  prep_ea<<<(E * NCH + 255) / 256, 256, 0, stream>>>(edge_attr, W_edge, b_edge, ea_lin, E);
  prep_v3<<<(E * NCH + 255) / 256, 256, 0, stream>>>(ea_lin, W_conv, v3, E);

  edge_gemm_ln_scatter<<<E, 256, 0, stream>>>(x, edge_idx, v3, Wf,
                                              ln_gamma, ln_beta, out, E);
}